// MultiModalPromptLearner_13700945674859
// MI455X (gfx1250) — compile-verified
//
#include <hip/hip_runtime.h>
#include <math.h>

// ---------------------------------------------------------------------------
// Problem constants (from reference)
// ---------------------------------------------------------------------------
constexpr int kHALF = 12;          // tokens per prompt
constexpr int kD    = 12;          // depth
constexpr int kDT   = 512;
constexpr int kDI   = 768;
constexpr int kE    = 1280;        // concat dim
constexpr int kH    = 80;          // hidden
constexpr int kKC   = 128;         // GEMM1 K-chunk staged in LDS
// LDS row strides padded so stride%32==4 -> row-indexed lane access hits
// distinct banks; stride*4 % 8 == 0 keeps b64 pair loads aligned.
constexpr int kXST  = 1284;        // x buffer row stride (floats)
constexpr int kHST  = 164;         // h buffer row stride (floats)
constexpr int kAST  = kKC + 4;     // a-chunk row stride (132)

// Workspace layout (float offsets). Variant tables, 16-row padded.
constexpr int OFF_IMG = 0;                             // [12][3][16][768]
constexpr int OFF_TXT = OFF_IMG + 12 * 3 * 16 * 768;   // 442368
constexpr int OFF_PCI = OFF_TXT + 12 * 3 * 16 * 512;   // 737280  [3][12][768]
constexpr int OFF_PCT = OFF_PCI + 3 * 12 * 768;        // 764928  [3][12][512]

typedef __attribute__((ext_vector_type(2))) float v2f;
typedef __attribute__((ext_vector_type(4))) float v4f;
typedef __attribute__((ext_vector_type(8))) float v8f;

__device__ __forceinline__ float gelu_exact(float x) {
    return 0.5f * x * (1.0f + erff(x * 0.70710678118654752440f));
}

// One V_WMMA_F32_16X16X4_F32 step, both operands from LDS.
// A 16x4: lane<16 -> row=l, K={k,k+1}; lane>=16 -> row=l, K={k+2,k+3}.
// B 4x16 mirrored: lane<16 -> col=l, K={k,k+1}; lane>=16 -> K={k+2,k+3}.
// ap: per-lane A row base (indexed by local k); bp: per-lane B column base.
__device__ __forceinline__ v8f wstep(const float* __restrict__ ap,
                                     const float* __restrict__ bp,
                                     int bstride, int kk, int hlf, v8f acc) {
    const int ka = kk + 2 * hlf;
    v2f a, b;
    a.x = ap[ka];
    a.y = ap[ka + 1];
    b.x = bp[ka * bstride];
    b.y = bp[(ka + 1) * bstride];
    return __builtin_amdgcn_wmma_f32_16x16x4_f32(false, a, false, b,
                                                 (short)0, acc, false, false);
}

// ---------------------------------------------------------------------------
// Kernel 1: full 12-depth MLP pipeline for the 3 missing_type variants.
// grid = 3 (variant), block = 512 (16 wave32s). State lives in LDS across
// depths; per-depth results spill to the ws variant tables. All WMMA operand
// traffic is LDS; global weight traffic is bulk coalesced staging.
// ---------------------------------------------------------------------------
__global__ __launch_bounds__(512) void pipeline_kernel(
    const float* __restrict__ vpc, const float* __restrict__ vpm,
    const float* __restrict__ tpc, const float* __restrict__ tpm,
    const float* __restrict__ lnig, const float* __restrict__ lnib,
    const float* __restrict__ lntg, const float* __restrict__ lntb,
    const float* __restrict__ iw1, const float* __restrict__ ib1,
    const float* __restrict__ iw2, const float* __restrict__ ib2,
    const float* __restrict__ tw1, const float* __restrict__ tb1,
    const float* __restrict__ tw2, const float* __restrict__ tb2,
    float* __restrict__ ws) {
    __shared__ float xbuf[16 * kXST];       // raw x (img|txt)        ~82 KB
    __shared__ float hbuf[16 * kHST];       // [16][img 80 | txt 80]  ~10 KB
    __shared__ float wbuf[kKC * 160];       // B slab (GEMM1: [128][160],
                                            //         GEMM2: [80][256]) 80 KB
    __shared__ float abuf[2 * 16 * kAST];   // LN'd A chunk           ~17 KB
    __shared__ float mstat[16], rstat[16];

    const int v    = blockIdx.x;        // variant == missing_type value
    const int tid  = threadIdx.x;
    const int w    = tid >> 5;          // wave id 0..15
    const int lane = tid & 31;
    const int hlf  = lane >> 4;
    const int l    = lane & 15;

    float* IMG_OUT = ws + OFF_IMG;
    float* TXT_OUT = ws + OFF_TXT;
    const float* imgsrc = (v == 2) ? vpm : vpc;
    const float* txtsrc = (v == 1) ? tpm : tpc;

    // init x = [img | txt], rows 12..15 zero
    for (int idx = tid; idx < 16 * kE; idx += 512) {
        const int r = idx / kE, c = idx - r * kE;
        float val = 0.0f;
        if (r < kHALF)
            val = (c < kDI) ? imgsrc[r * kDI + c] : txtsrc[r * kDT + (c - kDI)];
        xbuf[r * kXST + c] = val;
    }
    __syncthreads();

    for (int d = 0; d < kD; ++d) {
        // ---- (1) per-row LN stats (shared by both LN branches) ------------
        {
            const int row = w;  // 16 waves <-> 16 rows
            float s = 0.0f, q = 0.0f;
            for (int e = lane; e < kE; e += 32) {
                const float x = xbuf[row * kXST + e];
                s += x; q += x * x;
            }
            for (int off = 16; off; off >>= 1) {
                s += __shfl_down(s, off);
                q += __shfl_down(q, off);
            }
            if (lane == 0) {
                const float m   = s * (1.0f / kE);
                const float var = q * (1.0f / kE) - m * m;
                mstat[row] = m;
                rstat[row] = rsqrtf(var + 1e-5f);
            }
        }
        if (d + 1 < kD) {   // stream next depth's w1 toward L2
            const float* nw1 = iw1 + (size_t)(d + 1) * kE * kH;
            const float* nw2 = tw1 + (size_t)(d + 1) * kE * kH;
            for (int off = tid * 32; off < kE * kH; off += 512 * 32) {
                __builtin_prefetch(nw1 + off);
                __builtin_prefetch(nw2 + off);
            }
        }
        __syncthreads();

        // ---- (2) GEMM1 [16x1280]@[1280x80] x2 branches, K-chunked --------
        v8f a0 = {}, a1 = {}, a2 = {}, a3 = {};
        const int br1  = (w >= 5);                       // compute waves 0..9
        const int n0   = ((br1 ? (w - 5) : w) << 4);
        const float* ap1 = abuf + (br1 * 16 + l) * kAST; // A row base
        const float* bp1 = wbuf + br1 * kH + n0 + l;     // B column base

        for (int kc = 0; kc < kE; kc += kKC) {
            // stage w1 chunk (both branches) -> wbuf[k][0..159]
            for (int idx = tid; idx < kKC * kH; idx += 512) {
                const int kk = idx / kH, c = idx - kk * kH;
                const size_t rb = (size_t)(d * kE + kc + kk) * kH + c;
                wbuf[kk * 160 + c]      = iw1[rb];
                wbuf[kk * 160 + kH + c] = tw1[rb];
            }
            // stage LN'd A chunk: (x-m)*rstd*g + b for both branches
            for (int idx = tid; idx < 2 * 16 * kKC; idx += 512) {
                const int br = idx >> 11;            // 16*kKC = 2048
                const int rem = idx & 2047;
                const int r = rem >> 7, kk = rem & (kKC - 1);
                const int e = kc + kk;
                const float g = (br ? lntg : lnig)[d * kE + e];
                const float b = (br ? lntb : lnib)[d * kE + e];
                const float xn = (xbuf[r * kXST + e] - mstat[r]) * rstat[r];
                abuf[(br * 16 + r) * kAST + kk] = xn * g + b;
            }
            __syncthreads();
            if (w < 10) {                            // wave-uniform, EXEC=~0
                #pragma unroll
                for (int i = 0; i < kKC / 16; ++i) { // 4 indep acc chains
                    a0 = wstep(ap1, bp1, 160, 16 * i,      hlf, a0);
                    a1 = wstep(ap1, bp1, 160, 16 * i + 4,  hlf, a1);
                    a2 = wstep(ap1, bp1, 160, 16 * i + 8,  hlf, a2);
                    a3 = wstep(ap1, bp1, 160, 16 * i + 12, hlf, a3);
                }
            }
            __syncthreads();                         // wbuf/abuf reusable
        }
        if (w < 10) {                                // bias + exact GELU -> h
            const float* B1 = (br1 ? tb1 : ib1) + d * kH;
            const v8f acc = a0 + a1 + a2 + a3;
            #pragma unroll
            for (int j = 0; j < 8; ++j) {
                const int m = hlf * 8 + j;           // C/D: row = 8*half + j
                const float x = acc[j] + B1[n0 + l];
                hbuf[m * kHST + br1 * kH + n0 + l] = gelu_exact(x);
            }
        }
        __syncthreads();

        // ---- (3) GEMM2 [16x80]@[80x768|512], 256-col slabs per round -----
        for (int jj = 0; jj < 5; ++jj) {             // 0..2 img, 3..4 txt
            const int br    = (jj >= 3);
            const int cbase = (br ? (jj - 3) : jj) * 256;
            const int ld    = br ? kDT : kDI;
            const float* W2 = br ? (tw2 + (size_t)d * kH * kDT)
                                 : (iw2 + (size_t)d * kH * kDI);
            // stage [80][256] slab -> wbuf
            for (int idx = tid; idx < kH * 256; idx += 512) {
                const int kk = idx >> 8, c = idx & 255;
                wbuf[idx] = W2[kk * ld + cbase + c];
            }
            __syncthreads();
            {
                const float* ap2 = hbuf + l * kHST + br * kH;
                const float* bp2 = wbuf + (w << 4) + l;
                v8f c0 = {}, c1 = {}, c2 = {}, c3 = {};
                #pragma unroll
                for (int i = 0; i < kH / 16; ++i) {  // 20 WMMAs, 4 chains
                    c0 = wstep(ap2, bp2, 256, 16 * i,      hlf, c0);
                    c1 = wstep(ap2, bp2, 256, 16 * i + 4,  hlf, c1);
                    c2 = wstep(ap2, bp2, 256, 16 * i + 8,  hlf, c2);
                    c3 = wstep(ap2, bp2, 256, 16 * i + 12, hlf, c3);
                }
                const v8f acc = c0 + c1 + c2 + c3;
                const float* B2 = br ? (tb2 + d * kDT) : (ib2 + d * kDI);
                float* OUT = br ? (TXT_OUT + (size_t)((d * 3 + v) * 16) * kDT)
                                : (IMG_OUT + (size_t)((d * 3 + v) * 16) * kDI);
                const int n = cbase + (w << 4) + l;  // global column
                #pragma unroll
                for (int j = 0; j < 8; ++j) {
                    const int m = hlf * 8 + j;
                    const float val = acc[j] + B2[n];
                    OUT[m * ld + n] = val;
                    xbuf[m * kXST + br * kDI + n] = val;   // next depth's x
                }
            }
            __syncthreads();                         // wbuf free; xbuf slab done
        }
    }
}

// ---------------------------------------------------------------------------
// Kernel 2: common-prompt MLP for the 3 variants (tiny; VALU).
// ---------------------------------------------------------------------------
__global__ __launch_bounds__(256) void common_kernel(
    const float* __restrict__ cpc, const float* __restrict__ cpi,
    const float* __restrict__ cpt,
    const float* __restrict__ ciw1, const float* __restrict__ cib1,
    const float* __restrict__ ciw2, const float* __restrict__ cib2,
    const float* __restrict__ ctw1, const float* __restrict__ ctb1,
    const float* __restrict__ ctw2, const float* __restrict__ ctb2,
    float* __restrict__ ws) {
    __shared__ float xc[12 * 512];
    __shared__ float hcm[12 * 64];     // [row][img 32 | txt 32]
    const int v = blockIdx.x, tid = threadIdx.x;
    const float* src = (v == 0) ? cpc : (v == 1) ? cpi : cpt;
    float* PCIMG = ws + OFF_PCI;
    float* PCTXT = ws + OFF_PCT;

    for (int i = tid; i < 12 * 512; i += 256) xc[i] = src[i];
    __syncthreads();

    for (int o = tid; o < 768; o += 256) {       // 12*32 per branch
        const int br = (o >= 384);
        const int oo = o - br * 384;
        const int r = oo >> 5, c = oo & 31;
        const float* w1 = br ? ctw1 : ciw1;
        float s = (br ? ctb1 : cib1)[c];
        for (int k = 0; k < 512; ++k) s += xc[r * 512 + k] * w1[k * 32 + c];
        hcm[r * 64 + br * 32 + c] = gelu_exact(s);
    }
    __syncthreads();

    for (int o = tid; o < 12 * 768 + 12 * 512; o += 256) {
        if (o < 12 * 768) {
            const int r = o / 768, c = o - r * 768;
            float s = cib2[c];
            #pragma unroll
            for (int k = 0; k < 32; ++k) s += hcm[r * 64 + k] * ciw2[k * 768 + c];
            PCIMG[(v * 12 + r) * 768 + c] = s;
        } else {
            const int j = o - 12 * 768;
            const int r = j / 512, c = j - r * 512;
            float s = ctb2[c];
            #pragma unroll
            for (int k = 0; k < 32; ++k) s += hcm[r * 64 + 32 + k] * ctw2[k * 512 + c];
            PCTXT[(v * 12 + r) * 512 + c] = s;
        }
    }
}

// ---------------------------------------------------------------------------
// Kernel 3: bandwidth-bound gather/broadcast: variant tables -> d_out.
// float4 granularity, non-temporal stores (409 MB streamed once).
// ---------------------------------------------------------------------------
__global__ __launch_bounds__(256) void gather_kernel(
    const int* __restrict__ mt, const v4f* __restrict__ ws4,
    v4f* __restrict__ out) {
    constexpr int DI4 = kDI / 4, DT4 = kDT / 4;               // 192, 128
    constexpr int V1 = 512 * 24 * DI4;                        // img0
    constexpr int V2 = 512 * 24 * DT4;                        // txt0
    constexpr int PD3 = 512 * 12 * DI4;                       // per-depth imgs
    constexpr int PD4 = 512 * 12 * DT4;                       // per-depth txts
    constexpr int V3 = 11 * PD3;
    constexpr int V4 = 11 * PD4;
    constexpr int TOT = V1 + V2 + V3 + V4;
    constexpr int IMG4 = OFF_IMG / 4, TXT4 = OFF_TXT / 4;
    constexpr int PCI4 = OFF_PCI / 4, PCT4 = OFF_PCT / 4;

    const int stride = gridDim.x * blockDim.x;
    for (int i = blockIdx.x * blockDim.x + threadIdx.x; i < TOT; i += stride) {
        int src;
        if (i < V1) {                                          // img0 [B,24,768]
            const int b = i / (24 * DI4);
            const int rem = i - b * (24 * DI4);
            const int r = rem / DI4, c = rem - r * DI4;
            const int v = mt[b];
            src = (r < 12) ? (IMG4 + (v * 16 + r) * DI4 + c)           // d=0
                           : (PCI4 + (v * 12 + (r - 12)) * DI4 + c);
        } else if (i < V1 + V2) {                              // txt0 [B,24,512]
            const int j = i - V1;
            const int b = j / (24 * DT4);
            const int rem = j - b * (24 * DT4);
            const int r = rem / DT4, c = rem - r * DT4;
            const int v = mt[b];
            src = (r < 12) ? (TXT4 + (v * 16 + r) * DT4 + c)
                           : (PCT4 + (v * 12 + (r - 12)) * DT4 + c);
        } else if (i < V1 + V2 + V3) {                         // imgs[1:]
            const int j = i - (V1 + V2);
            const int d = j / PD3 + 1;
            const int rem = j - (d - 1) * PD3;
            const int b = rem / (12 * DI4);
            const int rem2 = rem - b * (12 * DI4);
            const int r = rem2 / DI4, c = rem2 - r * DI4;
            const int v = mt[b];
            src = IMG4 + ((d * 3 + v) * 16 + r) * DI4 + c;
        } else {                                               // txts[1:]
            const int j = i - (V1 + V2 + V3);
            const int d = j / PD4 + 1;
            const int rem = j - (d - 1) * PD4;
            const int b = rem / (12 * DT4);
            const int rem2 = rem - b * (12 * DT4);
            const int r = rem2 / DT4, c = rem2 - r * DT4;
            const int v = mt[b];
            src = TXT4 + ((d * 3 + v) * 16 + r) * DT4 + c;
        }
        const v4f val = ws4[src];
        __builtin_nontemporal_store(val, &out[i]);
    }
}

// ---------------------------------------------------------------------------
extern "C" void kernel_launch(void* const* d_in, const int* in_sizes, int n_in,
                              void* d_out, int out_size, void* d_ws,
                              size_t ws_size, hipStream_t stream) {
    (void)in_sizes; (void)n_in; (void)out_size; (void)ws_size;
    const int*   mt  = (const int*)d_in[0];
    const float* vpc = (const float*)d_in[1];
    const float* vpm = (const float*)d_in[2];
    const float* tpc = (const float*)d_in[3];
    const float* tpm = (const float*)d_in[4];
    const float* cpc = (const float*)d_in[5];
    const float* cpi = (const float*)d_in[6];
    const float* cpt = (const float*)d_in[7];
    const float* lnig = (const float*)d_in[8];
    const float* lnib = (const float*)d_in[9];
    const float* lntg = (const float*)d_in[10];
    const float* lntb = (const float*)d_in[11];
    const float* iw1 = (const float*)d_in[12];
    const float* ib1 = (const float*)d_in[13];
    const float* iw2 = (const float*)d_in[14];
    const float* ib2 = (const float*)d_in[15];
    const float* tw1 = (const float*)d_in[16];
    const float* tb1 = (const float*)d_in[17];
    const float* tw2 = (const float*)d_in[18];
    const float* tb2 = (const float*)d_in[19];
    const float* ciw1 = (const float*)d_in[20];
    const float* cib1 = (const float*)d_in[21];
    const float* ciw2 = (const float*)d_in[22];
    const float* cib2 = (const float*)d_in[23];
    const float* ctw1 = (const float*)d_in[24];
    const float* ctb1 = (const float*)d_in[25];
    const float* ctw2 = (const float*)d_in[26];
    const float* ctb2 = (const float*)d_in[27];
    float* ws = (float*)d_ws;

    pipeline_kernel<<<3, 512, 0, stream>>>(vpc, vpm, tpc, tpm,
                                           lnig, lnib, lntg, lntb,
                                           iw1, ib1, iw2, ib2,
                                           tw1, tb1, tw2, tb2, ws);
    common_kernel<<<3, 256, 0, stream>>>(cpc, cpi, cpt,
                                         ciw1, cib1, ciw2, cib2,
                                         ctw1, ctb1, ctw2, ctb2, ws);
    gather_kernel<<<8192, 256, 0, stream>>>(mt, (const v4f*)d_ws, (v4f*)d_out);
}